// PixelContrastLoss_37495064494136
// MI455X (gfx1250) — compile-verified
//
#include <hip/hip_runtime.h>

// PixelContrastLoss forward, fused flash-style (no 4096x4096 materialization).
// A = 32 anchors, V = 128 views, D = 512, N = A*V = 4096.
// feat[i] = X[(i%32)*128 + (i/32)] ; class(i) = y[i%32].

typedef __attribute__((ext_vector_type(2))) float v2f;
typedef __attribute__((ext_vector_type(8))) float v8f;

#define DIM          512
#define NROWS        4096
#define LDSA_STRIDE  516          // 512 + 4 pad -> conflict-free ds_read_b64
#define INV_T        (1.0f / 0.07f)

__device__ __forceinline__ float rowsum16(float v) {
    v += __shfl_xor(v, 1, 32);
    v += __shfl_xor(v, 2, 32);
    v += __shfl_xor(v, 4, 32);
    v += __shfl_xor(v, 8, 32);
    return v;
}

// log-sum-exp merge of per-lane (m, s) pairs across the 16-lane half-wave.
__device__ __forceinline__ void lse_merge16(float& m, float& s) {
#pragma unroll
    for (int step = 1; step <= 8; step <<= 1) {
        const float m2 = __shfl_xor(m, step, 32);
        const float s2 = __shfl_xor(s, step, 32);
        const float mn = fmaxf(m, m2);
        s = s * __expf(m - mn) + s2 * __expf(m2 - mn);
        m = mn;
    }
}

__device__ __forceinline__ size_t feat_base(int i) {
    // X flat index of feat row i: ((i%32)*128 + (i/32)) * 512
    return ((size_t)(i & 31) << 16) + ((size_t)(i >> 5) << 9);
}

__global__ __launch_bounds__(256) void pcl_main(const float* __restrict__ X,
                                                const int*   __restrict__ Y,
                                                float*       __restrict__ partial)
{
    __shared__ float ldsA[16 * LDSA_STRIDE];
    __shared__ float red_m[8][16];
    __shared__ float red_s[8][16];
    __shared__ float red_p[8][16];
    __shared__ float red_c[8][16];
    __shared__ int   cls[32];

    const int tid     = threadIdx.x;
    const int lane    = tid & 31;
    const int wave    = tid >> 5;
    const int lane15  = lane & 15;
    const int hi      = lane >> 4;      // 0: rows M=0..7, 1: rows M=8..15
    const int twohalf = hi * 2;         // K sub-offset inside each K=4 slice
    const int row_base = blockIdx.x * 16;

    if (tid < 32) cls[tid] = Y[tid];

    // ---- stage the 16x512 A-tile into LDS (padded rows) ----
    for (int idx = tid; idx < 16 * DIM; idx += 256) {
        const int r = idx >> 9;
        const int k = idx & (DIM - 1);
        ldsA[r * LDSA_STRIDE + k] = X[feat_base(row_base + r) + k];
    }
    __syncthreads();

    int rcls[8];
#pragma unroll
    for (int r = 0; r < 8; ++r) rcls[r] = cls[(row_base + r + hi * 8) & 31];

    const float* ap = &ldsA[lane15 * LDSA_STRIDE + twohalf];

    // per-LANE online softmax state (no cross-lane traffic in the hot loop)
    float mrow[8], negs[8];
#pragma unroll
    for (int r = 0; r < 8; ++r) { mrow[r] = -3.0e38f; negs[r] = 0.0f; }

    const int tile_lo = wave * 32;        // 256 column tiles split over 8 waves
    const int tile_hi = tile_lo + 32;

    // ================= pass 1: per-lane row max + negative exp-sum ========
    for (int t = tile_lo; t < tile_hi; ++t) {
        const int j = t * 16 + lane15;
        const float* bp = &X[feat_base(j) + twohalf];

        v8f acc0 = {}, acc1 = {};           // two chains -> more WMMA ILP
#pragma unroll 4
        for (int kk = 0; kk < DIM / 4; kk += 2) {
            v2f a0 = *(const v2f*)(ap + kk * 4);
            v2f b0 = *(const v2f*)(bp + kk * 4);
            acc0 = __builtin_amdgcn_wmma_f32_16x16x4_f32(
                       false, a0, false, b0, (short)0, acc0, false, false);
            v2f a1 = *(const v2f*)(ap + kk * 4 + 4);
            v2f b1 = *(const v2f*)(bp + kk * 4 + 4);
            acc1 = __builtin_amdgcn_wmma_f32_16x16x4_f32(
                       false, a1, false, b1, (short)0, acc1, false, false);
        }

        const int jc = cls[j & 31];
#pragma unroll
        for (int r = 0; r < 8; ++r) {
            const float l    = (acc0[r] + acc1[r]) * INV_T;
            const float mnew = fmaxf(mrow[r], l);
            const float e    = (rcls[r] == jc) ? 0.0f : __expf(l - mnew);
            negs[r] = negs[r] * __expf(mrow[r] - mnew) + e;
            mrow[r] = mnew;
        }
    }

    // ---- merge the 16 per-lane states into per-row states (once) ----
#pragma unroll
    for (int r = 0; r < 8; ++r) lse_merge16(mrow[r], negs[r]);

    // ---- merge per-wave (max, negsum) across the 8 waves ----
    if (lane15 == 0) {
#pragma unroll
        for (int r = 0; r < 8; ++r) {
            red_m[wave][r + hi * 8] = mrow[r];
            red_s[wave][r + hi * 8] = negs[r];
        }
    }
    __syncthreads();

    float mf[8], nf[8];
#pragma unroll
    for (int r = 0; r < 8; ++r) {
        const int M = r + hi * 8;
        float m = red_m[0][M];
#pragma unroll
        for (int w = 1; w < 8; ++w) m = fmaxf(m, red_m[w][M]);
        float s = 0.0f;
#pragma unroll
        for (int w = 0; w < 8; ++w) s += red_s[w][M] * __expf(red_m[w][M] - m);
        mf[r] = m;
        nf[r] = s;
    }

    // ================= pass 2: positive log-prob accumulation =============
    float psum[8], pcnt[8];
#pragma unroll
    for (int r = 0; r < 8; ++r) { psum[r] = 0.0f; pcnt[r] = 0.0f; }

    for (int t = tile_lo; t < tile_hi; ++t) {
        const int j = t * 16 + lane15;
        const float* bp = &X[feat_base(j) + twohalf];

        v8f acc0 = {}, acc1 = {};
#pragma unroll 4
        for (int kk = 0; kk < DIM / 4; kk += 2) {
            v2f a0 = *(const v2f*)(ap + kk * 4);
            v2f b0 = *(const v2f*)(bp + kk * 4);
            acc0 = __builtin_amdgcn_wmma_f32_16x16x4_f32(
                       false, a0, false, b0, (short)0, acc0, false, false);
            v2f a1 = *(const v2f*)(ap + kk * 4 + 4);
            v2f b1 = *(const v2f*)(bp + kk * 4 + 4);
            acc1 = __builtin_amdgcn_wmma_f32_16x16x4_f32(
                       false, a1, false, b1, (short)0, acc1, false, false);
        }

        const int jc = cls[j & 31];
#pragma unroll
        for (int r = 0; r < 8; ++r) {
            const int i = row_base + r + hi * 8;
            if (rcls[r] == jc && i != j) {
                const float l = (acc0[r] + acc1[r]) * INV_T - mf[r];
                psum[r] += l - __logf(__expf(l) + nf[r]);
                pcnt[r] += 1.0f;
            }
        }
    }

    // ---- reduce positives across lanes, then waves ----
#pragma unroll
    for (int r = 0; r < 8; ++r) {
        psum[r] = rowsum16(psum[r]);
        pcnt[r] = rowsum16(pcnt[r]);
    }
    if (lane15 == 0) {
#pragma unroll
        for (int r = 0; r < 8; ++r) {
            red_p[wave][r + hi * 8] = psum[r];
            red_c[wave][r + hi * 8] = pcnt[r];
        }
    }
    __syncthreads();

    if (wave == 0) {
        const int M = lane15;                 // duplicated in lanes 16..31; harmless
        float p = 0.0f, c = 0.0f;
#pragma unroll
        for (int w = 0; w < 8; ++w) { p += red_p[w][M]; c += red_c[w][M]; }
        float loss = -(p / (c + 1e-5f));      // T/BASE_T == 1
        loss = rowsum16(loss);                // sum of this block's 16 row losses
        if (lane == 0) partial[blockIdx.x] = loss;
    }
}

__global__ __launch_bounds__(256) void pcl_reduce(const float* __restrict__ partial,
                                                  float*       __restrict__ out)
{
    __shared__ float s[256];
    const int tid = threadIdx.x;
    s[tid] = partial[tid];
    __syncthreads();
    for (int off = 128; off >= 32; off >>= 1) {
        if (tid < off) s[tid] += s[tid + off];
        __syncthreads();
    }
    if (tid < 32) {
        float v = s[tid];
        v += __shfl_xor(v, 1, 32);
        v += __shfl_xor(v, 2, 32);
        v += __shfl_xor(v, 4, 32);
        v += __shfl_xor(v, 8, 32);
        v += __shfl_xor(v, 16, 32);
        if (tid == 0) out[0] = v / (float)NROWS;   // mean over 4096 rows
    }
}

extern "C" void kernel_launch(void* const* d_in, const int* in_sizes, int n_in,
                              void* d_out, int out_size, void* d_ws, size_t ws_size,
                              hipStream_t stream)
{
    const float* X = (const float*)d_in[0];   // (32, 128, 512) fp32
    const int*   Y = (const int*)d_in[1];     // (32,) int32
    float* ws  = (float*)d_ws;                // 256 block partials
    float* out = (float*)d_out;               // scalar loss

    pcl_main<<<dim3(NROWS / 16), dim3(256), 0, stream>>>(X, Y, ws);
    pcl_reduce<<<dim3(1), dim3(256), 0, stream>>>(ws, out);
}